// MultiHeadAttention_60447369724457
// MI455X (gfx1250) — compile-verified
//
#include <hip/hip_runtime.h>

typedef __attribute__((ext_vector_type(2))) float v2f;
typedef __attribute__((ext_vector_type(8))) float v8f;

#define NB 16
#define SS 1024
#define EE 256
#define HH 8
#define DD 32
#define E3 768
#define LDSTR 20

__device__ __forceinline__ v8f wmma_f32(v2f a, v2f b, v8f c) {
  // V_WMMA_F32_16X16X4_F32: D = A(16x4) * B(4x16) + C(16x16), fp32
  return __builtin_amdgcn_wmma_f32_16x16x4_f32(
      /*neg_a=*/false, a, /*neg_b=*/false, b,
      /*c_mod=*/(short)0, c, /*reuse_a=*/false, /*reuse_b=*/false);
}

// ---------------------------------------------------------------------------
// Strip GEMM: Y[M,N] = X[M,K] @ W[K,N] + bias[N].
// One wave per 64x16 output strip (4 stacked 16x16 C tiles). The 4 WMMAs per
// K-step share one B fragment, amortizing the strided W loads 4x; A loads are
// contiguous float2 (global_load_b64).
// ---------------------------------------------------------------------------
__global__ __launch_bounds__(256) void gemm64x16(const float* __restrict__ X,
                                                 const float* __restrict__ W,
                                                 const float* __restrict__ bias,
                                                 float* __restrict__ Y,
                                                 int M, int N, int K) {
  const int lane = threadIdx.x & 31;
  const int wave = threadIdx.x >> 5;
  const int ntn  = N >> 4;
  int strip = blockIdx.x * 8 + wave;
  int sm = strip / ntn;
  int tn = strip - sm * ntn;
  const int row0 = sm << 6, col0 = tn << 4;
  const int m = lane & 15, half = lane >> 4;

  // A fragment: lane = m + 16*(k>>1), vgpr = k&1 (f32 16x4 layout)
  const float* xrow = X + (size_t)(row0 + m) * K + 2 * half;
  // B fragment: lane = n, K = 2*half + j
  const float* wcol = W + (size_t)(2 * half) * N + col0 + m;

  v8f acc[4];
#pragma unroll
  for (int t = 0; t < 4; ++t) acc[t] = {};

#pragma unroll 4
  for (int kk = 0; kk < K; kk += 4) {
    v2f b;
    b.x = wcol[(size_t)kk * N];
    b.y = wcol[(size_t)(kk + 1) * N];
    v2f a0 = *(const v2f*)(xrow + kk);
    v2f a1 = *(const v2f*)(xrow + (size_t)16 * K + kk);
    v2f a2 = *(const v2f*)(xrow + (size_t)32 * K + kk);
    v2f a3 = *(const v2f*)(xrow + (size_t)48 * K + kk);
    acc[0] = wmma_f32(a0, b, acc[0]);
    acc[1] = wmma_f32(a1, b, acc[1]);
    acc[2] = wmma_f32(a2, b, acc[2]);
    acc[3] = wmma_f32(a3, b, acc[3]);
  }

  float bn = bias[col0 + m];
#pragma unroll
  for (int t = 0; t < 4; ++t) {
#pragma unroll
    for (int r = 0; r < 8; ++r) {
      int Mr = 16 * t + r + 8 * half;  // C layout: vgpr r -> row r+8*half
      Y[(size_t)(row0 + Mr) * N + col0 + m] = acc[t][r] + bn;
    }
  }
}

// ---------------------------------------------------------------------------
// In-place 2D RoPE on the q and k slices of the qkv buffer.
// ang[s][p] = (p<8 ? s/32 : s%32) * 10000^(-(p%8)/8), pairs (2p, 2p+1).
// ---------------------------------------------------------------------------
__global__ __launch_bounds__(256) void rope_kernel(float* __restrict__ qkv) {
  int t = blockIdx.x * 256 + threadIdx.x;
  int p = t & 15;  t >>= 4;   // pair index 0..15
  int h = t & 7;   t >>= 3;
  int s = t & 1023; t >>= 10;
  int b = t & 15;  t >>= 4;
  int which = t;              // 0 = q, 1 = k

  int i = s >> 5, j = s & 31;
  int r = p & 7;
  float freq = __expf(-1.1512925464970231f * (float)r);  // 10000^(-r/8)
  float pos  = (p < 8) ? (float)i : (float)j;
  float ang  = pos * freq;
  float sn, cs;
  __sincosf(ang, &sn, &cs);

  size_t idx = (size_t)(b * SS + s) * E3 + which * EE + h * DD + 2 * p;
  float x0 = qkv[idx], x1 = qkv[idx + 1];
  qkv[idx]     = x0 * cs - x1 * sn;
  qkv[idx + 1] = x0 * sn + x1 * cs;
}

// ---------------------------------------------------------------------------
// Flash attention: one wave per (b, h, 16-row q tile). Streams 64 key tiles
// with running max; the row-sum is kept as per-lane (per-column) partials
// rescaled by the uniform correction factor, and reduced once at the end --
// this removes 32 shuffle instructions per key tile vs naive online softmax.
// ---------------------------------------------------------------------------
__global__ __launch_bounds__(256) void attn_kernel(const float* __restrict__ qkv,
                                                   float* __restrict__ out) {
  __shared__ float lds_p[8][16 * LDSTR];

  const int lane = threadIdx.x & 31;
  const int wave = threadIdx.x >> 5;
  int id = blockIdx.x * 8 + wave;
  const int qt = id & 63; id >>= 6;
  const int h  = id & 7;  id >>= 3;
  const int b  = id;
  const int m = lane & 15, half = lane >> 4;
  const int row0 = qt << 4;

  const size_t basebh = (size_t)(b * SS) * E3 + h * DD;  // q slice base
  const float* kslab = qkv + basebh + EE;
  const float* vslab = qkv + basebh + 2 * EE;

  // Preload all 8 A-fragments of the 16x32 q tile (post-RoPE).
  v2f qa[8];
#pragma unroll
  for (int f = 0; f < 8; ++f)
    qa[f] = *(const v2f*)(qkv + basebh + (size_t)(row0 + m) * E3 + 4 * f + 2 * half);

  float mrow[8], lpart[8];
  v8f o0 = {}, o1 = {};
#pragma unroll
  for (int r = 0; r < 8; ++r) { mrow[r] = -3.0e38f; lpart[r] = 0.0f; }

  const float scale = 0.17677669529663687f;  // 1/sqrt(32)
  float* myp = &lds_p[wave][0];

  for (int kt = 0; kt < 64; ++kt) {
    const int krow0 = kt << 4;

    // S = q @ k^T : B[d][n] = k[krow0+n][d] -> contiguous float2 per lane
    v8f sacc = {};
    const float* kbase = kslab + (size_t)(krow0 + m) * E3 + 2 * half;
#pragma unroll
    for (int f = 0; f < 8; ++f) {
      v2f kb = *(const v2f*)(kbase + 4 * f);
      sacc = wmma_f32(qa[f], kb, sacc);
    }

    // Online max + deferred row-sum (corr is uniform across each half).
    float pr[8];
#pragma unroll
    for (int r = 0; r < 8; ++r) {
      float v = sacc[r] * scale;
      float rmax = v;
      rmax = fmaxf(rmax, __shfl_xor(rmax, 1, 32));
      rmax = fmaxf(rmax, __shfl_xor(rmax, 2, 32));
      rmax = fmaxf(rmax, __shfl_xor(rmax, 4, 32));
      rmax = fmaxf(rmax, __shfl_xor(rmax, 8, 32));
      float mnew = fmaxf(mrow[r], rmax);
      float corr = __expf(mrow[r] - mnew);
      float p    = __expf(v - mnew);
      lpart[r] = lpart[r] * corr + p;   // per-lane (per-column) partial sum
      mrow[r]  = mnew;
      o0[r] *= corr;
      o1[r] *= corr;
      pr[r] = p;
    }

    // C-layout -> A-layout for P via padded LDS (stride 20: conflict-free).
#pragma unroll
    for (int r = 0; r < 8; ++r)
      myp[(r + 8 * half) * LDSTR + m] = pr[r];
    __builtin_amdgcn_wave_barrier();
    v2f pa[4];
#pragma unroll
    for (int f = 0; f < 4; ++f)
      pa[f] = *(const v2f*)(myp + m * LDSTR + 4 * f + 2 * half);
    __builtin_amdgcn_wave_barrier();

    // O += P(16x16) @ V(16x32): two 16x16 C tiles across D.
#pragma unroll
    for (int f = 0; f < 4; ++f) {
      int kp = krow0 + 4 * f + 2 * half;
      const float* v0 = vslab + (size_t)kp * E3;
      const float* v1 = v0 + E3;
      v2f vb0, vb1;
      vb0.x = v0[m];      vb0.y = v1[m];
      vb1.x = v0[16 + m]; vb1.y = v1[16 + m];
      o0 = wmma_f32(pa[f], vb0, o0);
      o1 = wmma_f32(pa[f], vb1, o1);
    }
  }

  // Final row-sum reduction (once), normalize, scatter to (B, S, H*D).
#pragma unroll
  for (int r = 0; r < 8; ++r) {
    float lsum = lpart[r];
    lsum += __shfl_xor(lsum, 1, 32);
    lsum += __shfl_xor(lsum, 2, 32);
    lsum += __shfl_xor(lsum, 4, 32);
    lsum += __shfl_xor(lsum, 8, 32);
    float inv = 1.0f / lsum;
    int Mr = r + 8 * half;
    size_t orow = (size_t)(b * SS + row0 + Mr) * EE + h * DD;
    out[orow + m]      = o0[r] * inv;
    out[orow + 16 + m] = o1[r] * inv;
  }
}

// ---------------------------------------------------------------------------
extern "C" void kernel_launch(void* const* d_in, const int* in_sizes, int n_in,
                              void* d_out, int out_size, void* d_ws, size_t ws_size,
                              hipStream_t stream) {
  const float* x    = (const float*)d_in[0];
  const float* Wqkv = (const float*)d_in[1];
  const float* bqkv = (const float*)d_in[2];
  const float* Wout = (const float*)d_in[3];
  const float* bout = (const float*)d_in[4];
  float* out  = (float*)d_out;
  float* qkv  = (float*)d_ws;                       // 16384 x 768 fp32 (48 MB)
  float* attn = qkv + (size_t)NB * SS * E3;         // 16384 x 256 fp32 (16 MB)

  // 1) qkv = x @ Wqkv + bqkv   (256*48 strips / 8 waves = 1536 blocks)
  gemm64x16<<<dim3((NB * SS / 64) * (E3 / 16) / 8), dim3(256), 0, stream>>>(
      x, Wqkv, bqkv, qkv, NB * SS, E3, EE);

  // 2) RoPE in place on q and k slices (4M pairs / 256 = 16384 blocks)
  rope_kernel<<<dim3(2 * NB * SS * HH * (DD / 2) / 256), dim3(256), 0, stream>>>(qkv);

  // 3) flash attention (16*8*64 waves / 8 = 1024 blocks)
  attn_kernel<<<dim3(NB * HH * (SS / 16) / 8), dim3(256), 0, stream>>>(qkv, attn);

  // 4) out = attn @ Wout + bout (256*16 strips / 8 = 512 blocks)
  gemm64x16<<<dim3((NB * SS / 64) * (EE / 16) / 8), dim3(256), 0, stream>>>(
      attn, Wout, bout, out, NB * SS, EE, EE);
}